// XDGCNN_Cls_88905823027620
// MI455X (gfx1250) — compile-verified
//
#include <hip/hip_runtime.h>
#include <hip/hip_bf16.h>
#include <math.h>

// ---------------------------------------------------------------------------
// XDGCNN forward for MI455X (gfx1250, wave32). All GEMM-shaped compute runs
// through a generic strided-batched WMMA GEMM using V_WMMA_F32_16X16X4_F32
// (fp32 in / fp32 accumulate -> matches the fp32 reference numerics).
// GEMM wave computes a 32x32 C macro-tile (2x2 WMMA tiles) with clamped
// addressing (no EXEC divergence in the K-loop) and double-buffered loads.
// EdgeConv is factored through linearity:
//   max_k W.[x_j - x_i, x_i] = (Wb - Wa)x_i + max_k (Wa x_j)
// so each conv = 2 GEMMs + gather-max instead of materializing (b,n,k,2d).
// ---------------------------------------------------------------------------

typedef float v2f __attribute__((ext_vector_type(2)));
typedef float v8f __attribute__((ext_vector_type(8)));

#define EPS_BN 1e-5f
#define MAXK 16

__device__ __forceinline__ float gelu_exact(float v) {
    return 0.5f * v * (1.0f + erff(v * 0.70710678118654752f));
}

// ------------------------------- WMMA GEMM ---------------------------------
// C[z][m,n] = sum_k A[z][m,k] * B[z][k,n]  (+ bias[m]),  z = b*H + h
// Fully strided: handles transposes, weight sharing (stride 0), head batching.
struct GemmP {
    const float* A; const float* B; float* C; const float* bias;
    int M, N, K, H;
    long sAm, sAk, sAh, sAb;
    long sBk, sBn, sBh, sBb;
    long sCm, sCn, sCh, sCb;
};

__device__ __forceinline__ v8f wmma4(v2f a, v2f b, v8f c) {
    return __builtin_amdgcn_wmma_f32_16x16x4_f32(false, a, false, b,
                                                 (short)0, c, false, false);
}

__global__ __launch_bounds__(32) void gemm_wmma_f32(GemmP g) {
    const int tm = blockIdx.x * 32, tn = blockIdx.y * 32;
    const int z  = blockIdx.z;
    const int hz = z % g.H, bz = z / g.H;
    const float* A  = g.A + (long)hz * g.sAh + (long)bz * g.sAb;
    const float* Bm = g.B + (long)hz * g.sBh + (long)bz * g.sBb;
    float*       C  = g.C + (long)hz * g.sCh + (long)bz * g.sCb;

    const int lane = threadIdx.x & 31;
    const int r    = lane & 15;          // A row / B col within 16x16 tile
    const int kh   = (lane >> 4) << 1;   // K sub-offset: 0 (lanes 0-15) or 2
    const long sAm = g.sAm, sAk = g.sAk, sBk = g.sBk, sBn = g.sBn;

    // Clamp addresses instead of masking: out-of-range rows/cols compute
    // garbage that the bounds-checked store discards. Keeps the K-loop free
    // of EXEC divergence (WMMA requires EXEC all-ones anyway).
    const int am0 = min(tm + r,      g.M - 1);
    const int am1 = min(tm + 16 + r, g.M - 1);
    const int bn0 = min(tn + r,      g.N - 1);
    const int bn1 = min(tn + 16 + r, g.N - 1);

    const float* pa0 = A  + (long)am0 * sAm + (long)kh * sAk;
    const float* pa1 = A  + (long)am1 * sAm + (long)kh * sAk;
    const float* pb0 = Bm + (long)bn0 * sBn + (long)kh * sBk;
    const float* pb1 = Bm + (long)bn1 * sBn + (long)kh * sBk;
    const long aStep = 4 * sAk, bStep = 4 * sBk;

    v8f acc00 = {0.f,0.f,0.f,0.f,0.f,0.f,0.f,0.f};
    v8f acc01 = acc00, acc10 = acc00, acc11 = acc00;

    const int kFull = g.K & ~3;
    v2f a0, a1, b0, b1;
    if (kFull > 0) {                      // prologue: load first fragments
        a0.x = pa0[0]; a0.y = pa0[sAk];
        a1.x = pa1[0]; a1.y = pa1[sAk];
        b0.x = pb0[0]; b0.y = pb0[sBk];
        b1.x = pb1[0]; b1.y = pb1[sBk];
        pa0 += aStep; pa1 += aStep; pb0 += bStep; pb1 += bStep;
    }
    for (int k0 = 4; k0 < kFull; k0 += 4) {
        // double buffer: issue next loads before consuming current fragments
        v2f na0, na1, nb0, nb1;
        na0.x = pa0[0]; na0.y = pa0[sAk];
        na1.x = pa1[0]; na1.y = pa1[sAk];
        nb0.x = pb0[0]; nb0.y = pb0[sBk];
        nb1.x = pb1[0]; nb1.y = pb1[sBk];
        pa0 += aStep; pa1 += aStep; pb0 += bStep; pb1 += bStep;
        acc00 = wmma4(a0, b0, acc00);
        acc01 = wmma4(a0, b1, acc01);
        acc10 = wmma4(a1, b0, acc10);
        acc11 = wmma4(a1, b1, acc11);
        a0 = na0; a1 = na1; b0 = nb0; b1 = nb1;
    }
    if (kFull > 0) {
        acc00 = wmma4(a0, b0, acc00);
        acc01 = wmma4(a0, b1, acc01);
        acc10 = wmma4(a1, b0, acc10);
        acc11 = wmma4(a1, b1, acc11);
    }
    if (kFull < g.K) {                    // K tail (K % 4 != 0), guarded
        v2f ta0 = {0.f,0.f}, ta1 = {0.f,0.f}, tb0 = {0.f,0.f}, tb1 = {0.f,0.f};
        const int ka = kFull + kh;
        if (ka < g.K) {
            ta0.x = pa0[0]; ta1.x = pa1[0]; tb0.x = pb0[0]; tb1.x = pb1[0];
        }
        if (ka + 1 < g.K) {
            ta0.y = pa0[sAk]; ta1.y = pa1[sAk]; tb0.y = pb0[sBk]; tb1.y = pb1[sBk];
        }
        acc00 = wmma4(ta0, tb0, acc00);
        acc01 = wmma4(ta0, tb1, acc01);
        acc10 = wmma4(ta1, tb0, acc10);
        acc11 = wmma4(ta1, tb1, acc11);
    }

    // C 16x16 f32 fragment: VGPR i holds row (i + 8*(lane>=16)), col = lane&15
    const int cn0 = tn + (lane & 15);
    const int cn1 = cn0 + 16;
    const int mb  = (lane >> 4) << 3;
    const bool n0ok = cn0 < g.N, n1ok = cn1 < g.N;
#pragma unroll
    for (int i = 0; i < 8; ++i) {
        const int cm0 = tm + mb + i;
        const int cm1 = cm0 + 16;
        if (cm0 < g.M) {
            const float bs = g.bias ? g.bias[cm0] : 0.f;
            if (n0ok) C[(long)cm0 * g.sCm + (long)cn0 * g.sCn] = acc00[i] + bs;
            if (n1ok) C[(long)cm0 * g.sCm + (long)cn1 * g.sCn] = acc01[i] + bs;
        }
        if (cm1 < g.M) {
            const float bs = g.bias ? g.bias[cm1] : 0.f;
            if (n0ok) C[(long)cm1 * g.sCm + (long)cn0 * g.sCn] = acc10[i] + bs;
            if (n1ok) C[(long)cm1 * g.sCm + (long)cn1 * g.sCn] = acc11[i] + bs;
        }
    }
}

// --------------------------- elementwise kernels ---------------------------

// sqn[b,i] = sum_c x[b,c,i]^2   (x layout (B,D,N))
__global__ void rowsq_kernel(const float* x, float* sqn, int Bt, int D, int Np) {
    int t = blockIdx.x * blockDim.x + threadIdx.x;
    if (t >= Bt * Np) return;
    int b = t / Np, i = t - b * Np;
    const float* xb = x + (long)b * D * Np;
    float s = 0.f;
    for (int c = 0; c < D; ++c) { float v = xb[(long)c * Np + i]; s += v * v; }
    sqn[t] = s;
}

// k-NN (smallest distances; lowest-index tie-break like jax.lax.top_k).
// d2 computed expanded-form with max(.,0) clamp to match the reference.
__global__ void knn_kernel(const float* x, const float* sqn, int* idx,
                           int Bt, int D, int Np, int Kn) {
    int t = blockIdx.x * blockDim.x + threadIdx.x;
    if (t >= Bt * Np) return;
    int b = t / Np, i = t - b * Np;
    const float* xb = x + (long)b * D * Np;
    const float* sb = sqn + (long)b * Np;
    float xi[8];
    if (D <= 8) for (int c = 0; c < D; ++c) xi[c] = xb[(long)c * Np + i];
    float bd[MAXK]; int bi[MAXK];
    for (int p = 0; p < Kn; ++p) { bd[p] = 3.4e38f; bi[p] = 0; }
    const float si = sb[i];
    for (int j = 0; j < Np; ++j) {
        float dot = 0.f;
        if (D <= 8) { for (int c = 0; c < D; ++c) dot += xi[c] * xb[(long)c * Np + j]; }
        else        { for (int c = 0; c < D; ++c) dot += xb[(long)c * Np + i] * xb[(long)c * Np + j]; }
        float d2 = fmaxf(si + sb[j] - 2.f * dot, 0.f);
        if (d2 < bd[Kn - 1]) {
            int p = Kn - 1;
            while (p > 0 && d2 < bd[p - 1]) { bd[p] = bd[p - 1]; bi[p] = bi[p - 1]; --p; }
            bd[p] = d2; bi[p] = j;
        }
    }
    int* out = idx + ((long)b * Np + i) * Kn;
    for (int p = 0; p < Kn; ++p) out[p] = bi[p];
}

// out[b,c,i] = V[b,c,i] - U[b,c,i] + max_k U[b,c,idx[b,i,k]]
__global__ void gather_max_kernel(const float* U, const float* V, const int* idx,
                                  float* out, int Bt, int C, int Np, int Kn) {
    long t = (long)blockIdx.x * blockDim.x + threadIdx.x;
    long tot = (long)Bt * C * Np;
    if (t >= tot) return;
    int i = (int)(t % Np);
    int c = (int)((t / Np) % C);
    int b = (int)(t / ((long)Np * C));
    const float* Ub = U + ((long)b * C + c) * Np;
    const int*   ib = idx + ((long)b * Np + i) * Kn;
    float m = -3.4e38f;
    for (int k = 0; k < Kn; ++k) m = fmaxf(m, Ub[ib[k]]);
    out[t] = V[t] - Ub[i] + m;
}

// training-mode BatchNorm (+ optional residual) + exact GELU. One block per
// channel; safe to run in place (each element is touched by one thread).
__global__ __launch_bounds__(256) void bn_gelu_kernel(const float* xin, const float* res,
        const float* gamma, const float* beta, float* out, int Bt, int C, int Np) {
    __shared__ float s1[256], s2[256];
    const int c = blockIdx.x;
    const long cnt = (long)Bt * Np;
    float a1 = 0.f, a2 = 0.f;
    for (long t = threadIdx.x; t < cnt; t += 256) {
        long b = t / Np, i = t - b * Np;
        long off = ((long)b * C + c) * Np + i;
        float v = xin[off];
        if (res) v += res[off];
        a1 += v; a2 += v * v;
    }
    s1[threadIdx.x] = a1; s2[threadIdx.x] = a2;
    __syncthreads();
    for (int st = 128; st > 0; st >>= 1) {
        if ((int)threadIdx.x < st) { s1[threadIdx.x] += s1[threadIdx.x + st];
                                     s2[threadIdx.x] += s2[threadIdx.x + st]; }
        __syncthreads();
    }
    const float mean = s1[0] / (float)cnt;
    const float var  = s2[0] / (float)cnt - mean * mean;
    const float sc = gamma[c] * rsqrtf(var + EPS_BN);
    const float sh = beta[c];
    for (long t = threadIdx.x; t < cnt; t += 256) {
        long b = t / Np, i = t - b * Np;
        long off = ((long)b * C + c) * Np + i;
        float v = xin[off];
        if (res) v += res[off];
        out[off] = gelu_exact((v - mean) * sc + sh);
    }
}

// ChannelRMSNorm: x / max(||x||_c, 1e-12) * sqrt(C) * gamma[c]
__global__ void rmsnorm_kernel(const float* x, const float* gamma, float* out,
                               int Bt, int C, int Np) {
    int t = blockIdx.x * blockDim.x + threadIdx.x;
    if (t >= Bt * Np) return;
    int b = t / Np, i = t - b * Np;
    const float* xb = x + (long)b * C * Np + i;
    float s = 0.f;
    for (int c = 0; c < C; ++c) { float v = xb[(long)c * Np]; s += v * v; }
    const float sc = sqrtf((float)C) / fmaxf(sqrtf(s), 1e-12f);
    float* ob = out + (long)b * C * Np + i;
    for (int c = 0; c < C; ++c) ob[(long)c * Np] = xb[(long)c * Np] * sc * gamma[c];
}

// out[h,i] = sum_{d<Dh} x[(h*Dh+d)*Np + i]^2   (per-head column squared norms)
__global__ void colsq_kernel(const float* x, float* out, int H, int Dh, int Np) {
    int t = blockIdx.x * blockDim.x + threadIdx.x;
    if (t >= H * Np) return;
    int h = t / Np, i = t - h * Np;
    const float* xb = x + (long)h * Dh * Np + i;
    float s = 0.f;
    for (int d = 0; d < Dh; ++d) { float v = xb[(long)d * Np]; s += v * v; }
    out[t] = s;
}

// in-place:  qk[h,i,:] -> softmax_m( -sqrt(max(q2+k2-2*qk,0))/8 )
__global__ __launch_bounds__(256) void attn_softmax_kernel(float* qk, const float* q2,
        const float* k2, int n, int m) {
    __shared__ float red[256];
    const int i = blockIdx.x, h = blockIdx.y;
    float* row = qk + ((long)h * n + i) * m;
    const float qq = q2[h * n + i];
    const float* kk = k2 + (long)h * m;
    float mx = -3.4e38f;
    for (int j = threadIdx.x; j < m; j += 256) {
        float d2 = fmaxf(qq + kk[j] - 2.f * row[j], 0.f);
        float s = -sqrtf(d2) * 0.125f;   // * DIM_HEAD^-0.5
        row[j] = s;
        mx = fmaxf(mx, s);
    }
    red[threadIdx.x] = mx; __syncthreads();
    for (int st = 128; st > 0; st >>= 1) {
        if ((int)threadIdx.x < st) red[threadIdx.x] = fmaxf(red[threadIdx.x], red[threadIdx.x + st]);
        __syncthreads();
    }
    mx = red[0]; __syncthreads();
    float sm = 0.f;
    for (int j = threadIdx.x; j < m; j += 256) {
        float e = expf(row[j] - mx);
        row[j] = e; sm += e;
    }
    red[threadIdx.x] = sm; __syncthreads();
    for (int st = 128; st > 0; st >>= 1) {
        if ((int)threadIdx.x < st) red[threadIdx.x] += red[threadIdx.x + st];
        __syncthreads();
    }
    const float inv = 1.f / red[0];
    for (int j = threadIdx.x; j < m; j += 256) row[j] *= inv;
}

// out[b*C+c] = max_i x[b,c,i]
__global__ void maxpool_kernel(const float* x, float* out, int Bt, int C, int Np) {
    int t = blockIdx.x * blockDim.x + threadIdx.x;
    if (t >= Bt * C) return;
    const float* xb = x + (long)t * Np;
    float m = -3.4e38f;
    for (int i = 0; i < Np; ++i) m = fmaxf(m, xb[i]);
    out[t] = m;
}

// ------------------------------- host side ---------------------------------

static inline void gemm(hipStream_t s, const float* A, const float* B, float* C,
                        const float* bias, int M, int N, int K, int Z, int H,
                        long sAm, long sAk, long sAh, long sAb,
                        long sBk, long sBn, long sBh, long sBb,
                        long sCm, long sCn, long sCh, long sCb) {
    GemmP g{A, B, C, bias, M, N, K, H,
            sAm, sAk, sAh, sAb, sBk, sBn, sBh, sBb, sCm, sCn, sCh, sCb};
    dim3 grid((M + 31) / 32, (N + 31) / 32, Z);
    gemm_wmma_f32<<<grid, dim3(32), 0, s>>>(g);
}

extern "C" void kernel_launch(void* const* d_in, const int* in_sizes, int n_in,
                              void* d_out, int out_size, void* d_ws, size_t ws_size,
                              hipStream_t stream) {
    (void)in_sizes; (void)n_in; (void)out_size; (void)ws_size;
    const int B = 8, HEADS = 8, DH = 64;
    const int DIMS[4]   = {64, 128, 256, 512};
    const int NPTS[4]   = {256, 64, 16, 4};
    const int BLOCKS[4] = {3, 3, 9, 3};

    auto F = [&](int i) { return (const float*)d_in[i]; };
    // Flatten order: top-level dict insertion order (x, xyz, params);
    // params pytree in JAX sorted-key order:
    //  conv{b1,b2,g1,g2,w1,w2}, head_b, head_bias, head_g, head_w1, head_w2,
    //  mlp_b, mlp_beta, mlp_g, mlp_w,
    //  stages[i]: blocks[j]{b1,b2,g1,g2,w1,w2}..., down{attn{g_kv,g_q,wkv,wo,wq}, emb}
    const float* x_in  = F(0);   // (8,3,4096)
    const float* xyz   = F(1);   // (8,3,4096)
    const float* conv_b1 = F(2), *conv_b2 = F(3), *conv_g1 = F(4), *conv_g2 = F(5);
    const float* conv_w1 = F(6), *conv_w2 = F(7);
    const float* head_b = F(8), *head_bias = F(9), *head_g = F(10);
    const float* head_w1 = F(11), *head_w2 = F(12);
    const float* mlp_b = F(13), *mlp_beta = F(14), *mlp_g = F(15), *mlp_w = F(16);
    int stage_base[4];
    { int s = 17; for (int i = 0; i < 4; ++i) { stage_base[i] = s; s += 6 * BLOCKS[i] + 6; } }

    // ---- workspace carve (fp32 elements) ----
    float* w = (float*)d_ws;
    size_t o = 0;
    auto take = [&](size_t n) { float* p = w + o; o += n; return p; };
    float* bufA  = take(2u * 1024 * 1024);   // activations ping
    float* bufB  = take(2u * 1024 * 1024);   // activations pong
    float* bufU  = take(2u * 1024 * 1024);   // edgeconv U  / rms(emb)
    float* bufV  = take(2u * 1024 * 1024);   // edgeconv V
    float* bufR  = take(2u * 1024 * 1024);   // rms(context), full batch
    float* bufQ  = take(512u * 256);         // q (512 x n)
    float* bufQ2 = take(8u * 256);           // per-head ||q||^2
    float* bufK  = take(512u * 4096);        // per-b k
    float* bufVv = take(512u * 4096);        // per-b v
    float* bufK2 = take(8u * 4096);          // per-b ||k||^2 ; also knn sq-norms
    float* bufQK = take(8u * 256 * 4096);    // per-b scores ; also edgeconv h1 temp
    float* bufO  = take(512u * 256);         // per-b attn out (512 x n)
    float* bufP  = take(8u * 512);           // pooled features
    float* bufH1 = take(8u * 256);           // head hidden
    int*   bufIdx = (int*)take(8u * 4096 * 16);

    auto edgeconv = [&](float* h, const float* w1, const float* g1, const float* b1,
                        const float* w2, const float* g2, const float* b2,
                        int d, int np, int kn) {
        const long hb = (long)d * np;
        float* bufT = bufQK;  // h1 temp (free outside attention)
        // U = Wa h,  V = Wb h  (Wa = w1[:, :d], Wb = w1[:, d:], w1 row stride 2d)
        gemm(stream, w1,     h, bufU, nullptr, d, np, d, B, 1,
             2L * d, 1, 0, 0,  np, 1, 0, hb,  np, 1, 0, hb);
        gemm(stream, w1 + d, h, bufV, nullptr, d, np, d, B, 1,
             2L * d, 1, 0, 0,  np, 1, 0, hb,  np, 1, 0, hb);
        long tot = (long)B * d * np;
        gather_max_kernel<<<(tot + 255) / 256, 256, 0, stream>>>(bufU, bufV, bufIdx, bufT, B, d, np, kn);
        bn_gelu_kernel<<<d, 256, 0, stream>>>(bufT, nullptr, g1, b1, bufT, B, d, np);
        gemm(stream, w2,     bufT, bufU, nullptr, d, np, d, B, 1,
             2L * d, 1, 0, 0,  np, 1, 0, hb,  np, 1, 0, hb);
        gemm(stream, w2 + d, bufT, bufV, nullptr, d, np, d, B, 1,
             2L * d, 1, 0, 0,  np, 1, 0, hb,  np, 1, 0, hb);
        gather_max_kernel<<<(tot + 255) / 256, 256, 0, stream>>>(bufU, bufV, bufIdx, bufT, B, d, np, kn);
        bn_gelu_kernel<<<d, 256, 0, stream>>>(bufT, h /*residual*/, g2, b2, h /*in place*/, B, d, np);
    };

    // ---- stem: knn(xyz,16); h = gelu(bn(mlp_w @ x + mlp_b)); edgeconv ----
    {
        int np = 4096;
        rowsq_kernel<<<(B * np + 255) / 256, 256, 0, stream>>>(xyz, bufK2, B, 3, np);
        knn_kernel<<<(B * np + 127) / 128, 128, 0, stream>>>(xyz, bufK2, bufIdx, B, 3, np, 16);
        gemm(stream, mlp_w, x_in, bufB, mlp_b, 64, np, 3, B, 1,
             3, 1, 0, 0,  np, 1, 0, 3L * np,  np, 1, 0, 64L * np);
        bn_gelu_kernel<<<64, 256, 0, stream>>>(bufB, nullptr, mlp_g, mlp_beta, bufA, B, 64, np);
        edgeconv(bufA, conv_w1, conv_g1, conv_b1, conv_w2, conv_g2, conv_b2, 64, np, 16);
    }

    float* hcur = bufA;
    int cur_d = 64, cur_n = 4096;

    for (int si = 0; si < 4; ++si) {
        const int d = DIMS[si], np = NPTS[si];
        const int ctxd = cur_d, m = cur_n;
        const int ab = stage_base[si] + 6 * BLOCKS[si];
        const float* g_kv = F(ab + 0);
        const float* g_q  = F(ab + 1);
        const float* wkv  = F(ab + 2);   // (1024, ctxd)
        const float* wo   = F(ab + 3);   // (d, 512)
        const float* wq   = F(ab + 4);   // (512, d)
        const float* emb  = F(ab + 5);   // (1, d, np)
        float* hnext = (hcur == bufA) ? bufB : bufA;

        // ---- downsample = cross_attention(emb, hcur) ----
        rmsnorm_kernel<<<(np + 127) / 128, 128, 0, stream>>>(emb, g_q, bufU, 1, d, np);
        gemm(stream, wq, bufU, bufQ, nullptr, 512, np, d, 1, 1,
             d, 1, 0, 0,  np, 1, 0, 0,  np, 1, 0, 0);
        colsq_kernel<<<(HEADS * np + 127) / 128, 128, 0, stream>>>(bufQ, bufQ2, HEADS, DH, np);
        rmsnorm_kernel<<<(B * m + 127) / 128, 128, 0, stream>>>(hcur, g_kv, bufR, B, ctxd, m);
        for (int bb = 0; bb < B; ++bb) {
            const float* cb = bufR + (long)bb * ctxd * m;
            gemm(stream, wkv,               cb, bufK,  nullptr, 512, m, ctxd, 1, 1,
                 ctxd, 1, 0, 0,  m, 1, 0, 0,  m, 1, 0, 0);
            gemm(stream, wkv + 512L * ctxd, cb, bufVv, nullptr, 512, m, ctxd, 1, 1,
                 ctxd, 1, 0, 0,  m, 1, 0, 0,  m, 1, 0, 0);
            colsq_kernel<<<(HEADS * m + 127) / 128, 128, 0, stream>>>(bufK, bufK2, HEADS, DH, m);
            // scores[h,i,j] = sum_d q[h*64+d, i] * k[h*64+d, j]
            gemm(stream, bufQ, bufK, bufQK, nullptr, np, m, DH, HEADS, HEADS,
                 1, np, 64L * np, 0,   m, 1, 64L * m, 0,   m, 1, (long)np * m, 0);
            attn_softmax_kernel<<<dim3(np, HEADS), 256, 0, stream>>>(bufQK, bufQ2, bufK2, np, m);
            // out[h*64+d, i] = sum_j attn[h,i,j] * v[h*64+d, j]
            gemm(stream, bufQK, bufVv, bufO, nullptr, np, DH, m, HEADS, HEADS,
                 m, 1, (long)np * m, 0,   1, m, 64L * m, 0,   1, np, 64L * np, 0);
            gemm(stream, wo, bufO, hnext + (long)bb * d * np, nullptr, d, np, 512, 1, 1,
                 512, 1, 0, 0,  np, 1, 0, 0,  np, 1, 0, 0);
        }
        hcur = hnext; cur_d = d; cur_n = np;

        // ---- EdgeConv blocks (feature-space knn, k = 4) ----
        for (int bi = 0; bi < BLOCKS[si]; ++bi) {
            const int pb = stage_base[si] + 6 * bi;
            const float* b1 = F(pb + 0), *b2 = F(pb + 1);
            const float* g1 = F(pb + 2), *g2 = F(pb + 3);
            const float* w1 = F(pb + 4), *w2 = F(pb + 5);
            rowsq_kernel<<<(B * np + 127) / 128, 128, 0, stream>>>(hcur, bufK2, B, d, np);
            knn_kernel<<<(B * np + 127) / 128, 128, 0, stream>>>(hcur, bufK2, bufIdx, B, d, np, 4);
            edgeconv(hcur, w1, g1, b1, w2, g2, b2, d, np, 4);
        }
    }

    // ---- head: global max pool -> fc256 + BN + GELU -> fc40 + bias ----
    maxpool_kernel<<<(B * 512 + 255) / 256, 256, 0, stream>>>(hcur, bufP, B, 512, cur_n);
    gemm(stream, head_w1, bufP, bufH1, nullptr, 256, B, 512, 1, 1,
         512, 1, 0, 0,   1, 512, 0, 0,   1, 256, 0, 0);
    bn_gelu_kernel<<<256, 256, 0, stream>>>(bufH1, nullptr, head_g, head_b, bufH1, B, 256, 1);
    gemm(stream, head_w2, bufH1, (float*)d_out, head_bias, 40, B, 256, 1, 1,
         256, 1, 0, 0,   1, 256, 0, 0,   1, 40, 0, 0);
}